// RayleighNoise_23648089932393
// MI455X (gfx1250) — compile-verified
//
#include <hip/hip_runtime.h>
#include <cstdint>

#define BLOCK 256
#define MAXB  4096

typedef float v4f __attribute__((ext_vector_type(4)));

// ---- total-order float <-> uint mapping (for atomic min/max on floats) ----
__device__ __forceinline__ unsigned f2mono(float f) {
  unsigned u = __float_as_uint(f);
  return (u & 0x80000000u) ? ~u : (u | 0x80000000u);
}
__device__ __forceinline__ float mono2f(unsigned u) {
  return __uint_as_float((u & 0x80000000u) ? (u ^ 0x80000000u) : ~u);
}

__device__ __forceinline__ float rayleigh_noise(float uin) {
  // sigma * sqrt(-2 * log1p(-min(u, 1-1e-7)))
  float um = fminf(uin, 0.99999994f);   // largest float < 1.0
  float t  = 1.0f - um;                 // >= ~6e-8
  return 0.05f * __builtin_sqrtf(-2.0f * __logf(t));
}

__device__ __forceinline__ float clampf(float v, float lo, float hi) {
  return fminf(fmaxf(v, lo), hi);
}

// ---------------- init workspace (d_ws is poisoned; must init each call) ---
__global__ void init_ws(unsigned* ws) {
  ws[0] = 0xFFFFFFFFu;  // running min in mono space
  ws[1] = 0x00000000u;  // running max in mono space
}

// ---------------- pass 1: global min/max of x ------------------------------
__global__ void __launch_bounds__(BLOCK)
minmax_kernel(const float* __restrict__ x, int n4, long long n,
              unsigned* __restrict__ ws) {
  const v4f* x4 = (const v4f*)x;
  int gid    = blockIdx.x * blockDim.x + threadIdx.x;
  int stride = gridDim.x * blockDim.x;
  float lmin =  3.402823466e38f;
  float lmax = -3.402823466e38f;
  for (int i = gid; i < n4; i += stride) {
    v4f v = x4[i];  // regular temporal: warms L2 for pass 2
    lmin = fminf(lmin, fminf(fminf(v.x, v.y), fminf(v.z, v.w)));
    lmax = fmaxf(lmax, fmaxf(fmaxf(v.x, v.y), fmaxf(v.z, v.w)));
  }
  for (long long i = 4LL * n4 + gid; i < n; i += stride) {
    float v = x[i];
    lmin = fminf(lmin, v);
    lmax = fmaxf(lmax, v);
  }
  // wave32 butterfly reduction
#pragma unroll
  for (int off = 16; off > 0; off >>= 1) {
    lmin = fminf(lmin, __shfl_xor(lmin, off, 32));
    lmax = fmaxf(lmax, __shfl_xor(lmax, off, 32));
  }
  __shared__ float smin[BLOCK / 32], smax[BLOCK / 32];
  int lane = threadIdx.x & 31, wid = threadIdx.x >> 5;
  if (lane == 0) { smin[wid] = lmin; smax[wid] = lmax; }
  __syncthreads();
  if (threadIdx.x == 0) {
    float bmin = smin[0], bmax = smax[0];
#pragma unroll
    for (int i = 1; i < BLOCK / 32; ++i) {
      bmin = fminf(bmin, smin[i]);
      bmax = fmaxf(bmax, smax[i]);
    }
    atomicMin(&ws[0], f2mono(bmin));
    atomicMax(&ws[1], f2mono(bmax));
  }
}

// ---------------- pass 2 (bulk): async global->LDS double-buffered ---------
__global__ void __launch_bounds__(BLOCK)
noise_async(const float* __restrict__ x, const float* __restrict__ u,
            float* __restrict__ o, int ntiles, const unsigned* __restrict__ ws) {
  __shared__ v4f bx[2][BLOCK];
  __shared__ v4f bu[2][BLOCK];
  const float mn = mono2f(ws[0]);
  const float mx = mono2f(ws[1]);
  const int tid = threadIdx.x;
  const v4f* x4 = (const v4f*)x;
  const v4f* u4 = (const v4f*)u;
  v4f* o4 = (v4f*)o;

  const int step = gridDim.x;
  int linear = blockIdx.x;

  auto issue = [&](int lin, int buf) {
    // reversed traversal: start where pass 1 left x hot in the 192MB L2
    int t = ntiles - 1 - lin;
    size_t base = (size_t)t * BLOCK + tid;
    unsigned lx = (unsigned)(uintptr_t)&bx[buf][tid];  // generic->LDS: low 32 bits
    unsigned lu = (unsigned)(uintptr_t)&bu[buf][tid];
    // x: read twice overall -> regular temporal (exploit L2 reuse from pass 1)
    asm volatile("global_load_async_to_lds_b128 %0, %1, off"
                 :: "v"(lx), "v"((unsigned long long)(uintptr_t)(x4 + base))
                 : "memory");
    // u: streamed exactly once -> non-temporal, don't evict warm x from L2
    asm volatile("global_load_async_to_lds_b128 %0, %1, off th:TH_LOAD_NT"
                 :: "v"(lu), "v"((unsigned long long)(uintptr_t)(u4 + base))
                 : "memory");
  };

  if (linear < ntiles) issue(linear, 0);
  int buf = 0;
  for (; linear < ntiles; linear += step, buf ^= 1) {
    int nxt = linear + step;
    if (nxt < ntiles) {
      issue(nxt, buf ^ 1);                              // prefetch next tile
      asm volatile("s_wait_asynccnt 0x2" ::: "memory"); // current tile's 2 done
    } else {
      asm volatile("s_wait_asynccnt 0x0" ::: "memory");
    }
    v4f xv = bx[buf][tid];
    v4f uv = bu[buf][tid];
    v4f ov;
    ov.x = clampf(xv.x + rayleigh_noise(uv.x), mn, mx);
    ov.y = clampf(xv.y + rayleigh_noise(uv.y), mn, mx);
    ov.z = clampf(xv.z + rayleigh_noise(uv.z), mn, mx);
    ov.w = clampf(xv.w + rayleigh_noise(uv.w), mn, mx);
    int t = ntiles - 1 - linear;
    __builtin_nontemporal_store(ov, &o4[(size_t)t * BLOCK + tid]); // write-once: NT
  }
}

// ---------------- pass 2 (leftover float4 range) ---------------------------
__global__ void __launch_bounds__(BLOCK)
noise_direct(const float* __restrict__ x, const float* __restrict__ u,
             float* __restrict__ o, int start4, int end4,
             const unsigned* __restrict__ ws) {
  const float mn = mono2f(ws[0]);
  const float mx = mono2f(ws[1]);
  const v4f* x4 = (const v4f*)x;
  const v4f* u4 = (const v4f*)u;
  v4f* o4 = (v4f*)o;
  int gid    = blockIdx.x * blockDim.x + threadIdx.x;
  int stride = gridDim.x * blockDim.x;
  for (int i = start4 + gid; i < end4; i += stride) {
    v4f xv = x4[i];
    v4f uv = __builtin_nontemporal_load(&u4[i]);
    v4f ov;
    ov.x = clampf(xv.x + rayleigh_noise(uv.x), mn, mx);
    ov.y = clampf(xv.y + rayleigh_noise(uv.y), mn, mx);
    ov.z = clampf(xv.z + rayleigh_noise(uv.z), mn, mx);
    ov.w = clampf(xv.w + rayleigh_noise(uv.w), mn, mx);
    __builtin_nontemporal_store(ov, &o4[i]);
  }
}

// ---------------- pass 2 (scalar tail, n % 4) ------------------------------
__global__ void __launch_bounds__(BLOCK)
noise_tail(const float* __restrict__ x, const float* __restrict__ u,
           float* __restrict__ o, long long start, long long n,
           const unsigned* __restrict__ ws) {
  const float mn = mono2f(ws[0]);
  const float mx = mono2f(ws[1]);
  for (long long i = start + threadIdx.x; i < n; i += blockDim.x) {
    o[i] = clampf(x[i] + rayleigh_noise(u[i]), mn, mx);
  }
}

extern "C" void kernel_launch(void* const* d_in, const int* in_sizes, int n_in,
                              void* d_out, int out_size, void* d_ws, size_t ws_size,
                              hipStream_t stream) {
  const float* x = (const float*)d_in[0];
  const float* u = (const float*)d_in[1];
  float* o = (float*)d_out;
  unsigned* ws = (unsigned*)d_ws;

  long long n = (long long)in_sizes[0];
  int n4 = (int)(n >> 2);

  init_ws<<<1, 1, 0, stream>>>(ws);

  long long want = (n4 + BLOCK - 1) / BLOCK;
  int rblocks = (int)(want < 1 ? 1 : (want > MAXB ? MAXB : want));
  minmax_kernel<<<rblocks, BLOCK, 0, stream>>>(x, n4, n, ws);

  int ntiles = n4 / BLOCK;
  if (ntiles > 0) {
    int ablocks = ntiles < MAXB ? ntiles : MAXB;
    noise_async<<<ablocks, BLOCK, 0, stream>>>(x, u, o, ntiles, ws);
  }
  int start4 = ntiles * BLOCK;
  if (start4 < n4) {
    int cnt = n4 - start4;
    int dblocks = (cnt + BLOCK - 1) / BLOCK;
    if (dblocks > MAXB) dblocks = MAXB;
    noise_direct<<<dblocks, BLOCK, 0, stream>>>(x, u, o, start4, n4, ws);
  }
  long long start1 = 4LL * n4;
  if (start1 < n) {
    noise_tail<<<1, BLOCK, 0, stream>>>(x, u, o, start1, n, ws);
  }
}